// GNNModel_6081673691821
// MI455X (gfx1250) — compile-verified
//
#include <hip/hip_runtime.h>
#include <math.h>

typedef __attribute__((ext_vector_type(2))) float v2f;
typedef __attribute__((ext_vector_type(8))) float v8f;

#define D 128

// ---------------------------------------------------------------------------
// GEMM: out[N x 128] = act(A[N x 128] @ W[128 x 128] + bias), fp32 WMMA.
// Block = 256 threads = 8 waves; block covers 16 rows, wave w covers cols
// [16w, 16w+16). A strip staged in LDS (coalesced); W read from L2.
// ---------------------------------------------------------------------------
__global__ __launch_bounds__(256)
void gemm_bias_act_kernel(const float* __restrict__ A,
                          const float* __restrict__ W,
                          const float* __restrict__ bias,
                          float* __restrict__ out,
                          int relu)
{
    __shared__ float Al[16 * D];              // 8 KB
    const int m0  = blockIdx.x * 16;
    const int tid = threadIdx.x;

    // cooperative coalesced load of the 16x128 fp32 A strip
    {
        const float4* src = (const float4*)(A + (size_t)m0 * D);
        float4*       dst = (float4*)Al;
#pragma unroll
        for (int i = 0; i < (16 * D / 4) / 256; ++i)
            dst[tid + i * 256] = src[tid + i * 256];
    }
    __syncthreads();

    const int wave = tid >> 5;
    const int lane = tid & 31;
    const int n0   = wave * 16;
    const int half = lane >> 4;               // 0: lanes 0-15, 1: lanes 16-31
    const int l16  = lane & 15;
    const int koff = half * 2;                // A/B fragment K offset per half

    v8f c = {0.f, 0.f, 0.f, 0.f, 0.f, 0.f, 0.f, 0.f};
#pragma unroll
    for (int k0 = 0; k0 < D; k0 += 4) {
        v2f a, b;
        // A 16x4 fragment: M = l16; VGPR0 -> K = koff, VGPR1 -> K = koff+1
        a.x = Al[l16 * D + k0 + koff];
        a.y = Al[l16 * D + k0 + koff + 1];
        // B 4x16 fragment: N = n0+l16; VGPR0 -> K = koff, VGPR1 -> K = koff+1
        b.x = W[(size_t)(k0 + koff) * D + n0 + l16];
        b.y = W[(size_t)(k0 + koff + 1) * D + n0 + l16];
        c = __builtin_amdgcn_wmma_f32_16x16x4_f32(false, a, false, b,
                                                  (short)0, c, false, false);
    }

    // C/D layout: VGPR v -> row M = v + 8*half, col N = n0 + l16
    const int colIdx = n0 + l16;
    const float bv = bias[colIdx];
#pragma unroll
    for (int v = 0; v < 8; ++v) {
        const int rowIdx = m0 + v + half * 8;
        float r = c[v] + bv;
        if (relu) r = fmaxf(r, 0.0f);
        out[(size_t)rowIdx * D + colIdx] = r;
    }
}

// ---------------------------------------------------------------------------
// Utility kernels
// ---------------------------------------------------------------------------
__global__ void zero_kernel(float* __restrict__ p, size_t n)
{
    size_t i = (size_t)blockIdx.x * blockDim.x + threadIdx.x;
    if (i < n) p[i] = 0.0f;
}

__global__ void init_red_kernel(unsigned* __restrict__ red)
{
    if (threadIdx.x == 0) {
        red[0] = 0xFF800000u;        // -inf bit pattern (running max)
        ((float*)red)[1] = 0.0f;     // running sum
    }
}

__device__ inline void atomic_max_float_bits(unsigned* addr, float v)
{
    if (v >= 0.0f) atomicMax((int*)addr, __float_as_int(v));
    else           atomicMin(addr, (unsigned)__float_as_int(v));
}

// ---------------------------------------------------------------------------
// Edge scores: s[e] = leaky_relu(dot(h[row[e]], h[col[e]])), + global max.
// One wave per edge, one float4 per lane (coalesced 512B per node row).
// ---------------------------------------------------------------------------
__global__ __launch_bounds__(256)
void edge_score_kernel(const float* __restrict__ h,
                       const int* __restrict__ rowi,
                       const int* __restrict__ coli,
                       float* __restrict__ scores,
                       unsigned* __restrict__ redmax,
                       int E)
{
    __shared__ float smax[8];
    const int wave = threadIdx.x >> 5;
    const int lane = threadIdx.x & 31;
    const int e = blockIdx.x * 8 + wave;

    float s = -__builtin_inff();
    if (e < E) {
        const int r  = rowi[e];
        const int cl = coli[e];
        const float4 a = ((const float4*)(h + (size_t)r  * D))[lane];
        const float4 b = ((const float4*)(h + (size_t)cl * D))[lane];
        float d = a.x * b.x + a.y * b.y + a.z * b.z + a.w * b.w;
#pragma unroll
        for (int off = 16; off > 0; off >>= 1) d += __shfl_xor(d, off, 32);
        d = (d > 0.0f) ? d : 0.2f * d;       // leaky_relu, slope 0.2
        if (lane == 0) scores[e] = d;
        s = d;
    }
    if (lane == 0) smax[wave] = s;
    __syncthreads();
    if (threadIdx.x == 0) {
        float m = smax[0];
#pragma unroll
        for (int i = 1; i < 8; ++i) m = fmaxf(m, smax[i]);
        atomic_max_float_bits(redmax, m);
    }
}

// ---------------------------------------------------------------------------
// w[e] = exp(s[e] - max); accumulate global sum.
// ---------------------------------------------------------------------------
__global__ __launch_bounds__(256)
void edge_exp_kernel(float* __restrict__ scores,
                     const unsigned* __restrict__ redmax,
                     float* __restrict__ redsum,
                     int E)
{
    __shared__ float part[8];
    const int e = blockIdx.x * 256 + threadIdx.x;
    const float m = __int_as_float((int)redmax[0]);
    float w = 0.0f;
    if (e < E) {
        w = expf(scores[e] - m);
        scores[e] = w;
    }
#pragma unroll
    for (int off = 16; off > 0; off >>= 1) w += __shfl_xor(w, off, 32);
    const int lane = threadIdx.x & 31;
    const int wave = threadIdx.x >> 5;
    if (lane == 0) part[wave] = w;
    __syncthreads();
    if (threadIdx.x == 0) {
        float s = part[0];
#pragma unroll
        for (int i = 1; i < 8; ++i) s += part[i];
        atomicAdd(redsum, s);
    }
}

// ---------------------------------------------------------------------------
// Scatter: out[row[e]] += (w[e]/Z) * h[col[e]]. One wave per edge; the adds
// are L2-resident f32 atomics (25.6 MB destination fits in the 192 MB L2).
// ---------------------------------------------------------------------------
__global__ __launch_bounds__(256)
void edge_scatter_kernel(const float* __restrict__ h,
                         const int* __restrict__ rowi,
                         const int* __restrict__ coli,
                         const float* __restrict__ scores,
                         const float* __restrict__ sumptr,
                         float* __restrict__ out,
                         int E)
{
    const int wave = threadIdx.x >> 5;
    const int lane = threadIdx.x & 31;
    const int e = blockIdx.x * 8 + wave;
    if (e >= E) return;

    const float inv = 1.0f / sumptr[0];
    const float cfa = scores[e] * inv;
    const int r  = rowi[e];
    const int cl = coli[e];
    const float4 v = ((const float4*)(h + (size_t)cl * D))[lane];
    float* dst = out + (size_t)r * D + lane * 4;
    atomicAdd(dst + 0, cfa * v.x);
    atomicAdd(dst + 1, cfa * v.y);
    atomicAdd(dst + 2, cfa * v.z);
    atomicAdd(dst + 3, cfa * v.w);
}

// ---------------------------------------------------------------------------
// Launcher
// ---------------------------------------------------------------------------
extern "C" void kernel_launch(void* const* d_in, const int* in_sizes, int n_in,
                              void* d_out, int out_size, void* d_ws, size_t ws_size,
                              hipStream_t stream)
{
    const float* x  = (const float*)d_in[0];
    const int*   ei = (const int*)  d_in[1];
    const float* W0 = (const float*)d_in[2];
    const float* b0 = (const float*)d_in[3];
    const float* W1 = (const float*)d_in[4];
    const float* b1 = (const float*)d_in[5];
    const float* W2 = (const float*)d_in[6];
    const float* b2 = (const float*)d_in[7];
    float* out = (float*)d_out;

    const int N = in_sizes[0] / D;   // 50000 (divisible by 16)
    const int E = in_sizes[1] / 2;   // 800000
    const int* rowi = ei;
    const int* coli = ei + E;

    float* bufA   = (float*)d_ws;                    // features / GEMM out
    float* bufB   = bufA + (size_t)N * D;            // attention accumulator
    float* scores = bufB + (size_t)N * D;            // per-edge weights
    unsigned* red = (unsigned*)(scores + E);         // [0]=max bits, [1]=sum
    float* redsum = (float*)red + 1;

    const int    gemmGrid       = (N + 15) / 16;
    const int    edgeWaveGrid   = (E + 7) / 8;
    const int    edgeThreadGrid = (E + 255) / 256;
    const size_t nd             = (size_t)N * D;
    const int    zeroGrid       = (int)((nd + 255) / 256);

    // ----- layer 0: h = relu(x @ W0 + b0); attention -> bufB -----
    gemm_bias_act_kernel<<<gemmGrid, 256, 0, stream>>>(x, W0, b0, bufA, 1);
    zero_kernel<<<zeroGrid, 256, 0, stream>>>(bufB, nd);
    init_red_kernel<<<1, 32, 0, stream>>>(red);
    edge_score_kernel<<<edgeWaveGrid, 256, 0, stream>>>(bufA, rowi, coli, scores, red, E);
    edge_exp_kernel<<<edgeThreadGrid, 256, 0, stream>>>(scores, red, redsum, E);
    edge_scatter_kernel<<<edgeWaveGrid, 256, 0, stream>>>(bufA, rowi, coli, scores, redsum, bufB, E);

    // ----- layer 1: h = relu(bufB @ W1 + b1); attention -> bufB -----
    gemm_bias_act_kernel<<<gemmGrid, 256, 0, stream>>>(bufB, W1, b1, bufA, 1);
    zero_kernel<<<zeroGrid, 256, 0, stream>>>(bufB, nd);
    init_red_kernel<<<1, 32, 0, stream>>>(red);
    edge_score_kernel<<<edgeWaveGrid, 256, 0, stream>>>(bufA, rowi, coli, scores, red, E);
    edge_exp_kernel<<<edgeThreadGrid, 256, 0, stream>>>(scores, red, redsum, E);
    edge_scatter_kernel<<<edgeWaveGrid, 256, 0, stream>>>(bufA, rowi, coli, scores, redsum, bufB, E);

    // ----- output: out = bufB @ W2 + b2 (no relu) -----
    gemm_bias_act_kernel<<<gemmGrid, 256, 0, stream>>>(bufB, W2, b2, out, 0);
}